// PigeonRefiner_63617055589206
// MI455X (gfx1250) — compile-verified
//
#include <hip/hip_runtime.h>

// ---------------------------------------------------------------------------
// PigeonRefiner for MI455X (gfx1250, wave32).
//
// One wave32 per (b,k) pair:
//   1) scans proto_geocell for matching protos (ballot-compacted, ascending),
//   2) builds X = [embedding_b ; candidate proto means] (16 x 768) in LDS,
//   3) computes G = X * X^T with V_WMMA_F32_16X16X4_F32 (fp32, K-chunks of 4),
//      giving q2, dot(e,mean_j), ||mean_j||^2 in one Gram matrix,
//   4) repeats with X = [embedding_b ; members of best proto],
//   5) writes {best_dist, lat, lng, valid} to workspace.
// Pure global->LDS row copies use GLOBAL_LOAD_ASYNC_TO_LDS_B128 (ASYNCcnt)
// when the builtin is available; otherwise fall back to VALU copies.
// A second 32-thread kernel does the K=5 softmax / argmax / haversine epilogue.
// ---------------------------------------------------------------------------

typedef float v2f __attribute__((ext_vector_type(2)));
typedef float v8f __attribute__((ext_vector_type(8)));

#if __has_builtin(__builtin_amdgcn_global_load_async_to_lds_b128)
#define PIGEON_ASYNC_LDS 1
// Builtin parameter types (from clang diagnostic): 16-byte int vector pointers,
// global (addrspace(1), printed "__device__") source and LDS (addrspace(3))
// destination.
typedef int v4i_vs __attribute__((vector_size(4 * sizeof(int))));
typedef __attribute__((address_space(1))) v4i_vs* g_v4i_ptr;
typedef __attribute__((address_space(3))) v4i_vs* l_v4i_ptr;
#else
#define PIGEON_ASYNC_LDS 0
#endif

namespace {
constexpr int Dd   = 768;     // embedding dim
constexpr int Bq   = 32;      // batch
constexpr int Cc   = 10;      // candidate cells per query
constexpr int Kk   = 5;       // cells considered
constexpr int Pp   = 10000;   // protos
constexpr int Nn   = 100000;  // members
constexpr float TEMP    = 1.6f;
constexpr float MAX_REF = 1000.0f;
constexpr float EARTH_R = 6371.0f;
constexpr int MAXMATCH = 64;  // cap on protos per cell we track (data has 5)
}

__device__ __forceinline__ int lowerBoundI(const int* __restrict__ a, int n, int v) {
    int lo = 0, hi = n;
    while (lo < hi) { int mid = (lo + hi) >> 1; if (a[mid] < v) lo = mid + 1; else hi = mid; }
    return lo;
}
__device__ __forceinline__ int upperBoundI(const int* __restrict__ a, int n, int v) {
    int lo = 0, hi = n;
    while (lo < hi) { int mid = (lo + hi) >> 1; if (a[mid] <= v) lo = mid + 1; else hi = mid; }
    return lo;
}

// Async copy of one 768-float row global -> LDS (32 lanes x 6 chunks of 16B).
// Falls back to a VALU copy when the async builtin is unavailable.
__device__ __forceinline__ void copyRowToLds(float* __restrict__ dst,
                                             const float* __restrict__ src,
                                             int lane) {
#if PIGEON_ASYNC_LDS
    for (int t = lane; t < Dd / 4; t += 32) {
        __builtin_amdgcn_global_load_async_to_lds_b128(
            (g_v4i_ptr)(uintptr_t)(src + 4 * t),
            (l_v4i_ptr)(uintptr_t)(dst + 4 * t),
            /*offset=*/0, /*cpol=*/0);
    }
#else
    for (int d = lane; d < Dd; d += 32) dst[d] = src[d];
#endif
}

__device__ __forceinline__ void waitAsyncLds() {
#if PIGEON_ASYNC_LDS
#if __has_builtin(__builtin_amdgcn_s_wait_asynccnt)
    __builtin_amdgcn_s_wait_asynccnt(0);
#else
    asm volatile("s_wait_asynccnt 0" ::: "memory");
#endif
#endif
}

// Gram of the 16 x 768 LDS block X, via V_WMMA_F32_16X16X4_F32 chain.
// f32 A-fragment layout (ISA 7.12.2): lanes 0-15 -> M=lane, K={k0,k0+1};
// lanes 16-31 -> M=lane-16, K={k0+2,k0+3}. B 4x16 mirrors transposed, so the
// same fragment serves both operands when B-columns == A-rows (Gram).
// C/D layout: VGPR v: lanes 0-15 -> (M=v, N=lane); lanes 16-31 -> (M=v+8, N=lane-16).
__device__ __forceinline__ void gram16(const float* __restrict__ X,
                                       float* __restrict__ Gm, int lane) {
    const int half = lane >> 4;
    const int mrow = lane & 15;
    const float* rowp = X + mrow * Dd + 2 * half;
    v8f c = {};
    for (int k0 = 0; k0 < Dd; k0 += 4) {
        v2f f;
        f.x = rowp[k0];
        f.y = rowp[k0 + 1];
        c = __builtin_amdgcn_wmma_f32_16x16x4_f32(
                /*neg_a=*/false, f, /*neg_b=*/false, f,
                /*c_mod=*/(short)0, c, /*reuse_a=*/false, /*reuse_b=*/false);
    }
#pragma unroll
    for (int v = 0; v < 8; ++v)
        Gm[(v + 8 * half) * 16 + mrow] = c[v];
}

__global__ __launch_bounds__(32)
void pigeon_stage1(const float* __restrict__ embedding,
                   const int*   __restrict__ candidate_cells,
                   const float* __restrict__ emb_table,
                   const int*   __restrict__ member_img,
                   const int*   __restrict__ member_proto,
                   const int*   __restrict__ proto_geocell,
                   const float* __restrict__ proto_lat,
                   const float* __restrict__ proto_lng,
                   const float* __restrict__ img_lat,
                   const float* __restrict__ img_lng,
                   float*       __restrict__ slots) {
    __shared__ float X[16 * Dd];       // 48 KB row block
    __shared__ float Gm[256];          // 16x16 Gram
    __shared__ int   matched[MAXMATCH];
    __shared__ int   protoLoS[15];
    __shared__ int   protoCntS[15];

    const int wg   = blockIdx.x;
    const int b    = wg / Kk;
    const int k    = wg % Kk;
    const int lane = threadIdx.x;
    const int cell = candidate_cells[b * Cc + k];

    // row 0 = query embedding (persists through both WMMA stages); async copy
    // overlaps with the proto scan below.
    copyRowToLds(X, embedding + (size_t)b * Dd, lane);

    // ---- 1) collect matching protos in ascending order (argmin tie-break) ---
    int nM = 0;
    for (int base = 0; base < Pp; base += 32) {
        int p = base + lane;
        bool m = (p < Pp) && (proto_geocell[p] == cell);
        unsigned long long mk = __ballot(m);
        unsigned mk32 = (unsigned)mk;
        int ofs = __popc(mk32 & ((1u << lane) - 1u));
        if (m) {
            int pos = nM + ofs;
            if (pos < MAXMATCH) matched[pos] = p;
        }
        nM += __popc(mk32);
    }
    if (nM > MAXMATCH) nM = MAXMATCH;

    waitAsyncLds();
    __syncthreads();

    // ---- 2)+3) proto means -> Gram -> masked argmin over protos ------------
    float bestDist  = 3.0e38f;
    int   bestProto = -1, bestLo = 0, bestCnt = 0;

    for (int cbase = 0; cbase < nM; cbase += 15) {
        int cc = nM - cbase; if (cc > 15) cc = 15;
        for (int j = 0; j < cc; ++j) {
            int p   = matched[cbase + j];
            int lo  = lowerBoundI(member_proto, Nn, p);
            int hi  = upperBoundI(member_proto, Nn, p);
            int cnt = hi - lo;
            if (lane == 0) { protoLoS[j] = lo; protoCntS[j] = cnt; }
            // prefetch member rows (global_prefetch_b8)
            for (int m = lo + lane; m < hi; m += 32)
                __builtin_prefetch(emb_table + (size_t)member_img[m] * Dd, 0, 1);
            float inv = (cnt > 0) ? 1.0f / (float)cnt : 0.0f;
            for (int d = lane; d < Dd; d += 32) {
                float s = 0.0f;
                for (int m = lo; m < hi; ++m)
                    s += emb_table[(size_t)member_img[m] * Dd + d];
                X[(1 + j) * Dd + d] = s * inv;
            }
        }
        for (int j = cc; j < 15; ++j)
            for (int d = lane; d < Dd; d += 32) X[(1 + j) * Dd + d] = 0.0f;
        __syncthreads();

        gram16(X, Gm, lane);
        __syncthreads();

        float q2 = Gm[0];
        for (int j = 0; j < cc; ++j) {
            float dot  = Gm[1 + j];            // G[0][1+j] = <e, mean_j>
            float p2   = Gm[(1 + j) * 17];     // diagonal  = ||mean_j||^2
            float dist = sqrtf(fmaxf(q2 + p2 - 2.0f * dot, 0.0f));
            if (dist < bestDist) {             // strict <: first index wins
                bestDist  = dist;
                bestProto = matched[cbase + j];
                bestLo    = protoLoS[j];
                bestCnt   = protoCntS[j];
            }
        }
        __syncthreads();
    }

    // ---- 4) nearest member of the winning proto ----------------------------
    float bestMDist = 3.0e38f;
    int   bestImg   = -1;
    if (bestProto >= 0) {
        for (int mb = 0; mb < bestCnt; mb += 15) {
            int mm = bestCnt - mb; if (mm > 15) mm = 15;
            // pure copies: async global -> LDS (ASYNCcnt), no VGPR round-trip
            for (int i = 0; i < mm; ++i) {
                int img = member_img[bestLo + mb + i];
                copyRowToLds(&X[(1 + i) * Dd], emb_table + (size_t)img * Dd, lane);
            }
            for (int i = mm; i < 15; ++i)
                for (int d = lane; d < Dd; d += 32) X[(1 + i) * Dd + d] = 0.0f;
            waitAsyncLds();
            __syncthreads();

            gram16(X, Gm, lane);
            __syncthreads();

            float q2 = Gm[0];
            for (int i = 0; i < mm; ++i) {
                float dot  = Gm[1 + i];
                float m2   = Gm[(1 + i) * 17];
                float dist = sqrtf(fmaxf(q2 + m2 - 2.0f * dot, 0.0f));
                if (dist < bestMDist) {        // ascending member index order
                    bestMDist = dist;
                    bestImg   = member_img[bestLo + mb + i];
                }
            }
            __syncthreads();
        }
    }

    // ---- 5) emit per-(b,k) slot --------------------------------------------
    if (lane == 0) {
        float dist, lat, lng, valid;
        if (bestProto >= 0) {
            valid = 1.0f;
            dist  = bestDist;
            bool single = (bestCnt == 1);
            if (single || bestImg < 0) { lat = proto_lat[bestProto]; lng = proto_lng[bestProto]; }
            else                       { lat = img_lat[bestImg];     lng = img_lng[bestImg]; }
        } else {
            valid = 0.0f; dist = 3.0e38f; lat = 0.0f; lng = 0.0f;
        }
        float* s = slots + (size_t)(b * Kk + k) * 4;
        s[0] = dist; s[1] = lat; s[2] = lng; s[3] = valid;
    }
}

__global__ __launch_bounds__(32)
void pigeon_stage2(const float* __restrict__ initial_preds,
                   const int*   __restrict__ candidate_cells,
                   const float* __restrict__ candidate_probs,
                   const float* __restrict__ slots,
                   float*       __restrict__ out) {
    int b = threadIdx.x;
    if (b >= Bq) return;

    float sc[Kk], lat[Kk], lng[Kk];
#pragma unroll
    for (int k = 0; k < Kk; ++k) {
        const float* s = slots + (size_t)(b * Kk + k) * 4;
        float dist = s[0];
        lat[k] = s[1];
        lng[k] = s[2];
        sc[k]  = (s[3] > 0.5f) ? -dist : -100000.0f;
    }
    float mx = sc[0];
#pragma unroll
    for (int k = 1; k < Kk; ++k) mx = fmaxf(mx, sc[k]);
    float pr[Kk], se = 0.0f;
#pragma unroll
    for (int k = 0; k < Kk; ++k) { pr[k] = expf((sc[k] - mx) / TEMP); se += pr[k]; }
    int bi = 0; float bp = -1.0f;
#pragma unroll
    for (int k = 0; k < Kk; ++k) {
        float fp = candidate_probs[b * Cc + k] * (pr[k] / se);
        if (fp > bp) { bp = fp; bi = k; }   // strict >: first max wins
    }
    // haversine(initial_preds, chosen)
    const float RADF = 0.017453292519943295f;
    float p1 = initial_preds[b * 2]     * RADF;
    float p2 = lat[bi]                  * RADF;
    float dphi = p2 - p1;
    float dlmb = (lng[bi] - initial_preds[b * 2 + 1]) * RADF;
    float shp = sinf(dphi * 0.5f), shl = sinf(dlmb * 0.5f);
    float a = shp * shp + cosf(p1) * cosf(p2) * shl * shl;
    a = fminf(fmaxf(a, 0.0f), 1.0f);
    float dd = 2.0f * EARTH_R * asinf(sqrtf(a));
    if (dd > MAX_REF) bi = 0;

    out[b]           = lat[bi];
    out[Bq + b]      = lng[bi];
    out[2 * Bq + b]  = (float)candidate_cells[b * Cc + bi];
}

extern "C" void kernel_launch(void* const* d_in, const int* in_sizes, int n_in,
                              void* d_out, int out_size, void* d_ws, size_t ws_size,
                              hipStream_t stream) {
    const float* embedding       = (const float*)d_in[0];
    const float* initial_preds   = (const float*)d_in[1];
    const int*   candidate_cells = (const int*)  d_in[2];
    const float* candidate_probs = (const float*)d_in[3];
    const float* emb_table       = (const float*)d_in[4];
    const int*   member_img      = (const int*)  d_in[5];
    const int*   member_proto    = (const int*)  d_in[6];
    const int*   proto_geocell   = (const int*)  d_in[7];
    const float* proto_lat       = (const float*)d_in[8];
    const float* proto_lng       = (const float*)d_in[9];
    const float* img_lat         = (const float*)d_in[10];
    const float* img_lng         = (const float*)d_in[11];

    float* slots = (float*)d_ws;  // B*K*4 floats

    pigeon_stage1<<<dim3(Bq * Kk), dim3(32), 0, stream>>>(
        embedding, candidate_cells, emb_table, member_img, member_proto,
        proto_geocell, proto_lat, proto_lng, img_lat, img_lng, slots);

    pigeon_stage2<<<dim3(1), dim3(32), 0, stream>>>(
        initial_preds, candidate_cells, candidate_probs, slots, (float*)d_out);
}